// ModelNew_19688130085537
// MI455X (gfx1250) — compile-verified
//
#include <hip/hip_runtime.h>
#include <stdint.h>

// MI455X / gfx1250, wave32. Fused linear + folded eval-BatchNorm + (n+y)*y.
// B=1048576, DIN=64, DOUT=128, fp32 end-to-end via V_WMMA_F32_16X16X4_F32.
// x tiles staged to LDS with GLOBAL_LOAD_ASYNC_TO_LDS_B128 (ASYNCcnt),
// double-buffered so the DMA of tile m+1 overlaps the WMMA chain of tile m.
// Schedule pinned so ds loads + y loads issue before the 16-WMMA chain.

typedef float v2f __attribute__((ext_vector_type(2)));
typedef float v8f __attribute__((ext_vector_type(8)));

#define B_ROWS        1048576
#define DIN           64
#define DOUT          128
#define BLOCK_THREADS 256           // 8 waves (wave32)
#define MTILES        8             // 8 * 16 = 128 batch rows per block
#define ROWS_PER_BLK  (MTILES * 16)
#define LDS_STRIDE    68            // 64 floats + 4 pad -> conflict-free ds_load_2addr_b64

// Async DMA: memory -> LDS, 16B per lane, tracked by ASYNCcnt (no VGPR round-trip).
__device__ __forceinline__ void async_stage_b128(unsigned lds_off, const float* gptr) {
    asm volatile("global_load_async_to_lds_b128 %0, %1, off"
                 :: "v"(lds_off), "v"(gptr)
                 : "memory");
}
__device__ __forceinline__ void wait_async0() {
    asm volatile("s_wait_asynccnt 0x0" ::: "memory");
}

__global__ __launch_bounds__(BLOCK_THREADS)
void fused_linear_bn_mul_kernel(const float* __restrict__ x,
                                const float* __restrict__ y,
                                const float* __restrict__ W,
                                const float* __restrict__ bias,
                                const float* __restrict__ rmean,
                                const float* __restrict__ rvar,
                                const float* __restrict__ gamma,
                                const float* __restrict__ beta,
                                float* __restrict__ out) {
    // double-buffered x tile: 2 x (16 rows x 68 floats) = 8704 B of 320KB WGP LDS
    __shared__ float lds[2][16 * LDS_STRIDE];

    const int tid    = threadIdx.x;
    const int wave   = tid >> 5;        // 0..7  (wave32)
    const int lane   = tid & 31;
    const int laneLo = lane & 15;
    const int laneHi = lane >> 4;       // 0 or 1

    const int blockRow0 = (int)blockIdx.x * ROWS_PER_BLK;

    // Wave w owns output channels [16w, 16w+16); per-lane channel:
    const int col = wave * 16 + laneLo;

    // Fold eval-BN into per-channel scale/shift:
    //   normalized = lin*s + t,  s = gamma*rsqrt(var+eps),  t = (b - mean)*s + beta
    const float inv = rsqrtf(rvar[col] + 1e-5f);
    const float s_c = gamma[col] * inv;
    const float t_c = fmaf(bias[col] - rmean[col], s_c, beta[col]);

    // B-fragment: 64x16 slice of W^T for this wave (K=64, N=16), ISA layout:
    // VGPR v, lane L holds B[K = 4s + v + 2*(L>>4)][N = L&15] = W[col][K]
    v2f bfrag[16];
    const float* wrow = W + col * DIN;
#pragma unroll
    for (int s = 0; s < 16; ++s) {
        const int k = 4 * s + 2 * laneHi;
        bfrag[s] = *(const v2f*)(wrow + k);
    }

    // Staging geometry: 256 lanes x 16B = 4KB = one 16x64 x-tile.
    // LDS byte offset of this lane's 16B chunk (flat addr low 32 bits == LDS
    // address per ISA aperture rules).
    const int stage_r = tid >> 4;         // 0..15
    const int stage_c = (tid & 15) * 4;   // 0,4,...,60
    const unsigned lds_base  = (unsigned)(uintptr_t)(&lds[0][0]);
    const unsigned lds_elem  = (unsigned)((stage_r * LDS_STRIDE + stage_c) * 4);
    const unsigned buf_bytes = (unsigned)(16 * LDS_STRIDE * 4);
    const float*   gstage0   = x + (blockRow0 + stage_r) * DIN + stage_c;

    // Prologue: async-stage tile 0 into buffer 0
    async_stage_b128(lds_base + lds_elem, gstage0);

    for (int m = 0; m < MTILES; ++m) {
        const int cur = m & 1;

        // Tile m's DMA (issued last iteration / prologue) is the only
        // outstanding async op -> wait it, then barrier so every wave's
        // 512B chunk is visible before any ds_load.
        wait_async0();
        __syncthreads();

        // Issue DMA for tile m+1 into the other buffer. Safe: after the
        // barrier no wave still reads that buffer; overlaps the WMMAs below.
        if (m + 1 < MTILES) {
            async_stage_b128(lds_base + (cur ^ 1) * buf_bytes + lds_elem,
                             gstage0 + (m + 1) * 16 * DIN);
        }

        const int row0 = blockRow0 + m * 16;

        // Issue the streamed y loads for this tile's outputs first (longest
        // latency, independent of the matrix math).
        float yv[8];
#pragma unroll
        for (int d = 0; d < 8; ++d) {
            const int row = row0 + d + 8 * laneHi;
            yv[d] = __builtin_nontemporal_load(y + row * DOUT + col);
        }

        // Batch all A-fragment LDS reads (8x ds_load_2addr_b64).
        // A lane mapping: lane L holds A[M = L&15][K = 4s + v + 2*(L>>4)]
        const float* atile = &lds[cur][0];
        v2f afrag[16];
#pragma unroll
        for (int s = 0; s < 16; ++s) {
            const int k = 4 * s + 2 * laneHi;
            afrag[s] = *(const v2f*)(atile + laneLo * LDS_STRIDE + k);
        }

        // Keep all loads above, matrix chain below: no per-pair dscnt stalls.
        __builtin_amdgcn_sched_barrier(0);

        v8f acc = {};
#pragma unroll
        for (int s = 0; s < 16; ++s) {
            acc = __builtin_amdgcn_wmma_f32_16x16x4_f32(
                /*neg_a=*/false, afrag[s], /*neg_b=*/false, bfrag[s],
                /*c_mod=*/(short)0, acc, /*reuse_a=*/false, /*reuse_b=*/false);
        }

        // Epilogue: D-layout VGPR d, lane L -> out[M = d + 8*(L>>4)][N = L&15]
#pragma unroll
        for (int d = 0; d < 8; ++d) {
            const int row = row0 + d + 8 * laneHi;
            const float nv = fmaf(acc[d], s_c, t_c);
            __builtin_nontemporal_store((nv + yv[d]) * yv[d], out + row * DOUT + col);
        }
    }
}

extern "C" void kernel_launch(void* const* d_in, const int* in_sizes, int n_in,
                              void* d_out, int out_size, void* d_ws, size_t ws_size,
                              hipStream_t stream) {
    const float* x     = (const float*)d_in[0];
    const float* y     = (const float*)d_in[1];
    const float* W     = (const float*)d_in[2];
    const float* b     = (const float*)d_in[3];
    const float* rmean = (const float*)d_in[4];
    const float* rvar  = (const float*)d_in[5];
    const float* gamma = (const float*)d_in[6];
    const float* beta  = (const float*)d_in[7];
    float* out = (float*)d_out;

    dim3 grid(B_ROWS / ROWS_PER_BLK);   // 8192 blocks
    dim3 block(BLOCK_THREADS);
    hipLaunchKernelGGL(fused_linear_bn_mul_kernel, grid, block, 0, stream,
                       x, y, W, b, rmean, rvar, gamma, beta, out);
}